// DeepseekOcrImageTokenScatterBlock_25091198943419
// MI455X (gfx1250) — compile-verified
//
#include <hip/hip_runtime.h>
#include <stdint.h>

// Masked image-token scatter: 0 FLOPs, ~320 MiB streaming traffic ->
// HBM-bound (~14 us at 23.3 TB/s on MI455X). Strategy:
//   k1: mask prefix-sum in one workgroup. Specialized branch-free vector
//       kernel (b128 mask reads, VALU byte tests, int4 NT rank stores);
//       generic scalar kernel only as launch-time fallback.
//   k2: per-token row copy over the CDNA5 async global<->LDS engine
//       (global_load_async_to_lds_b128 / global_store_async_from_lds_b128,
//       ASYNCcnt) with NT temporal hints -- HBM->LDS->HBM, no VGPR staging.

typedef __attribute__((ext_vector_type(4))) unsigned int v4u;
typedef __attribute__((ext_vector_type(4))) int          v4i;

#define SCAN_THREADS 512  // 16 wave32s in one workgroup

// ---------------------------------------------------------------------------
// Shared scan core: Hillis-Steele inclusive scan over per-thread totals.
// ---------------------------------------------------------------------------
__device__ __forceinline__ int block_excl_prefix(int* sums, int tid, int total) {
  sums[tid] = total;
  __syncthreads();
  for (int off = 1; off < SCAN_THREADS; off <<= 1) {
    int v = (tid >= off) ? sums[tid - off] : 0;
    __syncthreads();
    sums[tid] += v;
    __syncthreads();
  }
  return (tid == 0) ? 0 : sums[tid - 1];
}

// ---------------------------------------------------------------------------
// Kernel 1a (fast path): requires ntok == SCAN_THREADS*items, items%16==0.
// No bounds checks anywhere -> straight-line b128 loads + VALU.
// ranks[t] = (# masked in [0..t]) - 1 if mask[t], else -1.
// ---------------------------------------------------------------------------
__global__ void __launch_bounds__(SCAN_THREADS)
rank_scan_vec_kernel(const uint8_t* __restrict__ mask,
                     int* __restrict__ ranks,
                     int nv) {  // 16-byte vectors per thread
  __shared__ int sums[SCAN_THREADS];
  const int tid = threadIdx.x;
  const v4u* mv = (const v4u*)(mask + (size_t)tid * (size_t)nv * 16u);

  int total = 0;
  for (int i = 0; i < nv; ++i) {
    v4u w = mv[i];
#pragma unroll
    for (int k = 0; k < 4; ++k) {
      unsigned x = w[k];
      total += ((x & 0x000000FFu) ? 1 : 0) + ((x & 0x0000FF00u) ? 1 : 0) +
               ((x & 0x00FF0000u) ? 1 : 0) + ((x & 0xFF000000u) ? 1 : 0);
    }
  }

  int run = block_excl_prefix(sums, tid, total);

  v4i* rv = (v4i*)(ranks + (size_t)tid * (size_t)nv * 16u);
  for (int i = 0; i < nv; ++i) {
    v4u w = mv[i];  // re-read: 16 KiB mask is cache-resident
#pragma unroll
    for (int k = 0; k < 4; ++k) {
      unsigned x = w[k];
      v4i r;
#pragma unroll
      for (int b = 0; b < 4; ++b) {
        const bool m = ((x >> (8 * b)) & 0xFFu) != 0;
        run += m ? 1 : 0;
        r[b] = m ? (run - 1) : -1;
      }
      __builtin_nontemporal_store(r, &rv[i * 4 + k]);
    }
  }
}

// ---------------------------------------------------------------------------
// Kernel 1b (generic fallback): any ntok / alignment.
// ---------------------------------------------------------------------------
__global__ void __launch_bounds__(SCAN_THREADS)
rank_scan_gen_kernel(const uint8_t* __restrict__ mask,
                     int* __restrict__ ranks,
                     int ntok, int items) {
  __shared__ int sums[SCAN_THREADS];
  const int tid  = threadIdx.x;
  const int base = tid * items;

  int total = 0;
  for (int i = 0; i < items; ++i) {
    const int idx = base + i;
    if (idx < ntok) total += (mask[idx] != 0) ? 1 : 0;
  }

  int run = block_excl_prefix(sums, tid, total);

  for (int i = 0; i < items; ++i) {
    const int idx = base + i;
    if (idx < ntok) {
      const bool m = (mask[idx] != 0);
      run += m ? 1 : 0;
      ranks[idx] = m ? (run - 1) : -1;
    }
  }
}

// ---------------------------------------------------------------------------
// CDNA5 async global<->LDS copy primitives (ASYNCcnt path), NT hints.
// LDS operand = wave-relative LDS byte address = low 32 bits of flat addr
// (ISA 10.2).
// ---------------------------------------------------------------------------
__device__ __forceinline__ void async_gl2lds_b128_nt(unsigned lds, uint64_t ga) {
  asm volatile("global_load_async_to_lds_b128 %0, %1, off th:TH_LOAD_NT"
               :: "v"(lds), "v"(ga) : "memory");
}
__device__ __forceinline__ void async_lds2gl_b128_nt(unsigned lds, uint64_t ga) {
  asm volatile("global_store_async_from_lds_b128 %1, %0, off th:TH_STORE_NT"
               :: "v"(lds), "v"(ga) : "memory");
}
__device__ __forceinline__ void wait_asynccnt0() {
  asm volatile("s_wait_asynccnt 0x0" ::: "memory");
}

// ---------------------------------------------------------------------------
// Kernel 2: one 256-thread block (8 wave32s) per token row.
// Each lane stages 16B chunks HBM -> LDS -> HBM via the async engine; the
// only sync is the per-wave s_wait_asynccnt between its loads and stores
// (each lane's store reads exactly the LDS bytes its own load wrote).
// ---------------------------------------------------------------------------
__global__ void __launch_bounds__(256)
scatter_copy_async_kernel(const float* __restrict__ embeds,
                          const float* __restrict__ images,
                          const int* __restrict__ ranks,
                          float* __restrict__ out,
                          int H) {
  extern __shared__ char stage[];  // H*4 bytes of staging
  const int token = blockIdx.x;
  const int r = ranks[token];      // block-uniform -> s_load_b32

  const float* src_row = (r >= 0) ? (images + (size_t)r * (size_t)H)
                                  : (embeds + (size_t)token * (size_t)H);
  float* dst_row = out + (size_t)token * (size_t)H;

  const uint64_t srcg  = (uint64_t)(uintptr_t)src_row;
  const uint64_t dstg  = (uint64_t)(uintptr_t)dst_row;
  const unsigned lbase = (unsigned)(uintptr_t)stage;  // LDS byte offset

  const int tid    = threadIdx.x;
  const int nchunk = H >> 2;       // 16B chunks per row (H=2048 -> 512)

  for (int i = tid; i < nchunk; i += 256) {
    async_gl2lds_b128_nt(lbase + (unsigned)i * 16u, srcg + (uint64_t)i * 16u);
  }
  wait_asynccnt0();                // this wave's LDS writes have landed
  for (int i = tid; i < nchunk; i += 256) {
    async_lds2gl_b128_nt(lbase + (unsigned)i * 16u, dstg + (uint64_t)i * 16u);
  }
  // S_ENDPGM performs an implicit wait-idle; no trailing wait needed.

  // Tail floats if H is not a multiple of 4 (not hit for H=2048).
  for (int i = (nchunk << 2) + tid; i < H; i += 256) {
    dst_row[i] = src_row[i];
  }
}

// ---------------------------------------------------------------------------
// Launch: dims derived from in_sizes.
//   d_in[0] = inputs_embeds  [B,S,H] f32
//   d_in[1] = images_seq_mask[B,S]   bool (1 byte/elem)
//   d_in[2] = images_in_this_batch [N_IMG,H] f32
// ---------------------------------------------------------------------------
extern "C" void kernel_launch(void* const* d_in, const int* in_sizes, int n_in,
                              void* d_out, int out_size, void* d_ws, size_t ws_size,
                              hipStream_t stream) {
  const float*   embeds = (const float*)d_in[0];
  const uint8_t* mask   = (const uint8_t*)d_in[1];
  const float*   images = (const float*)d_in[2];
  float*         out    = (float*)d_out;

  const int ntok = in_sizes[1];          // B*S
  const int H    = in_sizes[0] / ntok;   // hidden size

  int* ranks = (int*)d_ws;               // ntok ints of scratch

  const int items = (ntok + SCAN_THREADS - 1) / SCAN_THREADS;
  if ((items % 16 == 0) && (ntok == SCAN_THREADS * items)) {
    rank_scan_vec_kernel<<<1, SCAN_THREADS, 0, stream>>>(mask, ranks, items / 16);
  } else {
    rank_scan_gen_kernel<<<1, SCAN_THREADS, 0, stream>>>(mask, ranks, ntok, items);
  }

  const size_t lds_bytes = (size_t)H * sizeof(float);
  scatter_copy_async_kernel<<<ntok, 256, lds_bytes, stream>>>(
      embeds, images, ranks, out, H);
}